// HyperZZW_L_38577396253147
// MI455X (gfx1250) — compile-verified
//
#include <hip/hip_runtime.h>
#include <hip/hip_bf16.h>

// ---------------------------------------------------------------------------
// HyperZZW_L on MI455X (gfx1250, wave32)
//
// Fusion: pool(conv1x1(x)) == conv1x1(pool(x))  (both linear, disjoint axes)
//  K1: avg-pool x1 -> pooledT[392][64]              (reads 25.7 MB)
//  K2: WMMA fp32 GEMM (392x64x64) + slow_b          (tiny; v_wmma_f32_16x16x4_f32)
//  K3: depthwise dynamic 7x7 conv of x2 + bias      (reads 25.7MB, writes 25.7MB)
// Memory-bound: ~77 MB total -> ~3.3us floor at 23.3 TB/s.
// ---------------------------------------------------------------------------

typedef __attribute__((ext_vector_type(2))) float v2f;
typedef __attribute__((ext_vector_type(8))) float v8f;

#define CC     64
#define HH     112
#define WW     112
#define IMG    (HH * WW)          // 12544
#define NPIX   49                 // 7x7 dynamic kernel
#define MROWS  392                // 8 * 49 pooled rows
#define MPAD   400                // padded to 25 tiles of 16
#define MTILES 25

// ---------------------------------------------------------------------------
// Kernel 1: adaptive avg-pool 112x112 -> 7x7 (16x16 means), transposed layout
// pooledT[p][c], p = b*49 + ky*7 + kx.  One block per (b,c); thread (ty,tx)
// owns one element of each 16x16 region; wave32 shuffle + LDS reduce.
// ---------------------------------------------------------------------------
__global__ void pool16x16_kernel(const float* __restrict__ x1,
                                 float* __restrict__ pooledT) {
  const int bc = blockIdx.x;            // 0..511
  const int b  = bc >> 6;
  const int c  = bc & 63;
  const int tid = threadIdx.x;          // 0..255
  const int ty = tid >> 4, tx = tid & 15;

  __shared__ float red[8];
  const float* img = x1 + (size_t)bc * IMG;

  for (int q = 0; q < NPIX; ++q) {
    const int ky = q / 7, kx = q % 7;
    float v = img[(ky * 16 + ty) * WW + kx * 16 + tx];
    // wave32 reduction
    #pragma unroll
    for (int off = 16; off > 0; off >>= 1)
      v += __shfl_down(v, off, 32);
    if ((tid & 31) == 0) red[tid >> 5] = v;
    __syncthreads();
    if (tid == 0) {
      float s = 0.f;
      #pragma unroll
      for (int w = 0; w < 8; ++w) s += red[w];
      pooledT[(size_t)(b * NPIX + q) * CC + c] = s * (1.0f / 256.0f);
    }
    __syncthreads();
  }
}

// ---------------------------------------------------------------------------
// Kernel 2: D[p][d] = sum_c pooledT[p][c] * slow_w[d][c] + slow_b[d]
// One wave (32 threads) per 16x16 output tile; 16 steps of
// v_wmma_f32_16x16x4_f32 over K=64.
//
// ISA 7.12.2 32-bit layouts (wave32):
//   A (16x4):  VGPRv holds K=v (lanes 0-15, M=lane) / K=v+2 (lanes 16-31)
//   B (4x16):  VGPRv holds row K=v (lanes 0-15, N=lane) / K=v+2 (lanes 16-31)
//   C/D(16x16):VGPRv holds M=v (lanes 0-15) / M=v+8 (lanes 16-31), N=lane&15
// Per lane both A and B are a float2 of consecutive K -> plain b64 loads.
// B[k][n] = slow_w[n][k], i.e. read slow_w row-major per output channel n.
//
// A-loads are UNCONDITIONAL: pooled rows 392..399 are allocated workspace
// (uninitialized); row m of A only influences row m of D, and those D rows
// are masked out at the store. This keeps EXEC constant through the k-loop
// so all 32 global_load_b64 issue up front and the WMMA chain pipelines.
// ---------------------------------------------------------------------------
__global__ void slow_gemm_wmma_kernel(const float* __restrict__ pooledT,
                                      const float* __restrict__ slow_w,
                                      const float* __restrict__ slow_b,
                                      float* __restrict__ kernelBuf) {
  const int mt = blockIdx.x >> 2;         // 0..24
  const int nt = blockIdx.x & 3;          // 0..3
  const int m_base = mt * 16;
  const int n_base = nt * 16;

  const int lane = threadIdx.x;           // 0..31
  const int lo = lane & 15;
  const int hi = lane >> 4;               // 0 or 1

  const int arow = m_base + lo;           // pooled pixel index (0..399, unguarded)
  const int ncol = n_base + lo;           // output channel for B / store

  const float* aptr = pooledT + (size_t)arow * CC + hi * 2;
  const float* bptr = slow_w  + (size_t)ncol * CC + hi * 2;

  v8f acc = {};
  #pragma unroll
  for (int kk = 0; kk < 16; ++kk) {
    v2f a  = *(const v2f*)(aptr + kk * 4);
    v2f bm = *(const v2f*)(bptr + kk * 4);
    acc = __builtin_amdgcn_wmma_f32_16x16x4_f32(
        /*neg_a=*/false, a, /*neg_b=*/false, bm,
        /*c_mod=*/(short)0, acc, /*reuse_a=*/false, /*reuse_b=*/false);
  }

  const float bias = slow_b[ncol];
  #pragma unroll
  for (int v = 0; v < 8; ++v) {
    const int p = m_base + v + hi * 8;    // pooled pixel (row of D)
    if (p < MROWS) {
      const int bb = p / NPIX, q = p % NPIX;
      kernelBuf[(size_t)(bb * CC + ncol) * NPIX + q] = acc[v] + bias;
    }
  }
}

// ---------------------------------------------------------------------------
// Kernel 3: depthwise dynamic 7x7 conv, SAME zero padding, + fast_bias.
// One block per (b,c, 28-row tile). 34x120 LDS tile with halo, 49 weights in
// registers, 4-wide output strips with 10-float sliding window.
// ---------------------------------------------------------------------------
#define TY    28
#define TROWS (TY + 6)          // 34
#define TPITCH 120              // 112 + 6 halo, padded to mult of 4

__global__ void __launch_bounds__(256)
dyn_dw_conv7_kernel(const float* __restrict__ x2,
                    const float* __restrict__ kernelBuf,
                    const float* __restrict__ fast_bias,
                    float* __restrict__ out) {
  const int bc  = blockIdx.x >> 2;        // 0..511
  const int ty0 = (blockIdx.x & 3) * TY;  // 0,28,56,84
  const int tid = threadIdx.x;

  __shared__ __align__(16) float tile[TROWS * TPITCH];
  __shared__ float wsh[NPIX];

  if (tid < NPIX) wsh[tid] = kernelBuf[(size_t)bc * NPIX + tid];

  const float* img = x2 + (size_t)bc * IMG;
  for (int i = tid; i < TROWS * TPITCH; i += 256) {
    const int r  = i / TPITCH;
    const int cc = i - r * TPITCH;
    const int gy = ty0 + r - 3;
    const int gx = cc - 3;
    float v = 0.f;
    if (gy >= 0 && gy < HH && gx >= 0 && gx < WW) v = img[gy * WW + gx];
    tile[i] = v;
  }
  __syncthreads();

  const float bias = fast_bias[0];
  float wr[NPIX];
  #pragma unroll
  for (int i = 0; i < NPIX; ++i) wr[i] = wsh[i];

  float* dst = out + (size_t)bc * IMG;
  for (int g = tid; g < TY * 28; g += 256) {
    const int oy  = g / 28;               // 0..27
    const int ox4 = (g - oy * 28) * 4;    // 0..108 step 4
    float a0 = bias, a1 = bias, a2 = bias, a3 = bias;
    #pragma unroll
    for (int ky = 0; ky < 7; ++ky) {
      const float* row = &tile[(oy + ky) * TPITCH + ox4];
      const float4 r0 = *(const float4*)(row);      // cols 0..3  (16B aligned)
      const float4 r1 = *(const float4*)(row + 4);  // cols 4..7
      const float2 r2 = *(const float2*)(row + 8);  // cols 8..9
      const float win[10] = {r0.x, r0.y, r0.z, r0.w,
                             r1.x, r1.y, r1.z, r1.w,
                             r2.x, r2.y};
      #pragma unroll
      for (int kx = 0; kx < 7; ++kx) {
        const float wk = wr[ky * 7 + kx];
        a0 = fmaf(wk, win[kx + 0], a0);
        a1 = fmaf(wk, win[kx + 1], a1);
        a2 = fmaf(wk, win[kx + 2], a2);
        a3 = fmaf(wk, win[kx + 3], a3);
      }
    }
    float4 o = {a0, a1, a2, a3};
    *(float4*)(dst + (ty0 + oy) * WW + ox4) = o;
  }
}

// ---------------------------------------------------------------------------
extern "C" void kernel_launch(void* const* d_in, const int* in_sizes, int n_in,
                              void* d_out, int out_size, void* d_ws, size_t ws_size,
                              hipStream_t stream) {
  (void)in_sizes; (void)n_in; (void)out_size; (void)ws_size;
  const float* x1        = (const float*)d_in[0];
  const float* x2        = (const float*)d_in[1];
  const float* slow_w    = (const float*)d_in[2];
  const float* slow_b    = (const float*)d_in[3];
  const float* fast_bias = (const float*)d_in[4];
  float* out = (float*)d_out;

  // workspace layout: pooledT padded to 400 rows so the GEMM can load rows
  // 392..399 unconditionally; kernelBuf follows.
  float* pooledT   = (float*)d_ws;                                    // 400*64
  float* kernelBuf = (float*)((char*)d_ws + MPAD * CC * sizeof(float)); // 8*64*49

  pool16x16_kernel<<<dim3(8 * CC), dim3(256), 0, stream>>>(x1, pooledT);
  slow_gemm_wmma_kernel<<<dim3(MTILES * 4), dim3(32), 0, stream>>>(
      pooledT, slow_w, slow_b, kernelBuf);
  dyn_dw_conv7_kernel<<<dim3(8 * CC * 4), dim3(256), 0, stream>>>(
      x2, kernelBuf, fast_bias, out);
}